// Model_87058987090447
// MI455X (gfx1250) — compile-verified
//
// CDNA5 / gfx1250 implementation of the reference forecasting model.
// All GEMMs (incl. rFFT-as-DFT-matmul) go through a generic batched WMMA
// kernel using V_WMMA_F32_16X16X4_F32 (fp32, matching reference precision;
// the model is latency/bandwidth bound so low-precision WMMA buys nothing).
// GEMM v2: branchless edge guards (clamped addresses + selects, EXEC stays
// all-ones), uniform full-tile fast path, 16x32 tile per wave (2 accumulators
// sharing the A fragment), global_prefetch of the next K slab.
#include <hip/hip_runtime.h>
#include <math.h>

typedef float v2f __attribute__((ext_vector_type(2)));
typedef float v8f __attribute__((ext_vector_type(8)));

constexpr int BB  = 8;      // batch
constexpr int LL  = 336;    // input length
constexpr int PP  = 336;    // pred length
constexpr int CC  = 321;    // channels
constexpr int FF  = 169;    // rfft bins (L/2+1)
constexpr int DD  = 512;    // D
constexpr int IDD = 64;     // ID
constexpr int CLD = 64;     // CL
constexpr int GG  = 64;     // G
constexpr int GD  = 640;    // GDIM
constexpr int BC  = BB * CC;

static inline unsigned ceil_div(long n, long d) { return (unsigned)((n + d - 1) / d); }

__device__ __forceinline__ v8f wmma_f32(v2f a, v2f b, v8f c)
{
    return __builtin_amdgcn_wmma_f32_16x16x4_f32(false, a, false, b, (short)0, c,
                                                 false, false);
}

// ---------------------------------------------------------------------------
// Generic batched WMMA GEMM:  C[b] = act( op(A[b]) * op(B[b]) + bias[b] (+C) )
// One wave (32 threads) per 16x32 output tile (two 16x16 accumulators that
// share the A fragment); K loop in steps of 4 using V_WMMA_F32_16X16X4_F32.
// A 16x4 layout (ISA 7.12.2): lane L holds row=L&15; lanes 0-15 -> K,K+1;
// lanes 16-31 -> K+2,K+3.  B 4x16 mirrored.  C/D: acc[i] is row i + 8*(L>=16),
// col = L&15.  Edge tiles use clamped in-bounds addresses + value selects, so
// loads are unconditional and EXEC is never touched (WMMA requires EXEC=~0).
// ACT: 0=none 1=tanh 2=relu 3=adjacency-mask ((m==n)?-1e9:max(v,0))
// ---------------------------------------------------------------------------
template <bool TRANSA, bool TRANSB, int ACT, bool ACCUM, bool HASBIAS>
__global__ void gemm_wmma_kernel(const float* __restrict__ A, int lda, long long sA,
                                 const float* __restrict__ B, int ldb, long long sB,
                                 const float* __restrict__ bias, long long sBias,
                                 float* __restrict__ Cm, int ldc, long long sC,
                                 int M, int N, int K)
{
    const int batch = blockIdx.z;
    A  += (long long)batch * sA;
    B  += (long long)batch * sB;
    Cm += (long long)batch * sC;
    const float* bp = HASBIAS ? (bias + (long long)batch * sBias) : nullptr;

    const int n0    = blockIdx.x * 32;
    const int m0    = blockIdx.y * 16;
    const int lane  = threadIdx.x;        // 0..31 (wave32)
    const int l15   = lane & 15;
    const int khalf = (lane >> 4) << 1;   // 0 or 2
    const int arow  = m0 + l15;
    const int bcol0 = n0 + l15;
    const int bcol1 = bcol0 + 16;

    v8f acc0 = {0.f, 0.f, 0.f, 0.f, 0.f, 0.f, 0.f, 0.f};
    v8f acc1 = acc0;

    const int  K4   = K & ~3;
    const bool full = (m0 + 16 <= M) && (n0 + 32 <= N);   // scalar-uniform

    if (full) {
        // ---- fast path: no guards at all ----
        for (int k = 0; k < K4; k += 4) {
            const int ka = k + khalf;
            v2f a, b0, b1;
            if (TRANSA) {
                a.x = A[(long)ka * lda + arow];
                a.y = A[(long)(ka + 1) * lda + arow];
                __builtin_prefetch(&A[(long)(ka + 16) * lda + arow], 0, 3);
            } else {
                a.x = A[(long)arow * lda + ka];
                a.y = A[(long)arow * lda + ka + 1];
                __builtin_prefetch(&A[(long)arow * lda + ka + 64], 0, 3);
            }
            if (TRANSB) {
                b0.x = B[(long)bcol0 * ldb + ka];
                b0.y = B[(long)bcol0 * ldb + ka + 1];
                b1.x = B[(long)bcol1 * ldb + ka];
                b1.y = B[(long)bcol1 * ldb + ka + 1];
            } else {
                b0.x = B[(long)ka * ldb + bcol0];
                b0.y = B[(long)(ka + 1) * ldb + bcol0];
                b1.x = B[(long)ka * ldb + bcol1];
                b1.y = B[(long)(ka + 1) * ldb + bcol1];
            }
            acc0 = wmma_f32(a, b0, acc0);
            acc1 = wmma_f32(a, b1, acc1);
        }
    } else {
        // ---- edge path: clamped addresses + selects (branchless) ----
        const bool av  = arow  < M;
        const bool bv0 = bcol0 < N;
        const bool bv1 = bcol1 < N;
        const int  ar  = av  ? arow  : (M - 1);
        const int  bc0 = bv0 ? bcol0 : (N - 1);
        const int  bc1 = bv1 ? bcol1 : (N - 1);
        for (int k = 0; k < K4; k += 4) {
            const int ka = k + khalf;                 // ka, ka+1 < K4 <= K
            v2f a, b0, b1;
            float t0, t1;
            t0 = TRANSA ? A[(long)ka * lda + ar] : A[(long)ar * lda + ka];
            t1 = TRANSA ? A[(long)(ka + 1) * lda + ar] : A[(long)ar * lda + ka + 1];
            a.x = av ? t0 : 0.f;  a.y = av ? t1 : 0.f;
            t0 = TRANSB ? B[(long)bc0 * ldb + ka] : B[(long)ka * ldb + bc0];
            t1 = TRANSB ? B[(long)bc0 * ldb + ka + 1] : B[(long)(ka + 1) * ldb + bc0];
            b0.x = bv0 ? t0 : 0.f;  b0.y = bv0 ? t1 : 0.f;
            t0 = TRANSB ? B[(long)bc1 * ldb + ka] : B[(long)ka * ldb + bc1];
            t1 = TRANSB ? B[(long)bc1 * ldb + ka + 1] : B[(long)(ka + 1) * ldb + bc1];
            b1.x = bv1 ? t0 : 0.f;  b1.y = bv1 ? t1 : 0.f;
            acc0 = wmma_f32(a, b0, acc0);
            acc1 = wmma_f32(a, b1, acc1);
        }
    }

    // ---- K remainder (K % 4 != 0): one clamped step, valid for both paths ----
    if (K4 < K) {
        const bool av  = arow  < M;
        const bool bv0 = bcol0 < N;
        const bool bv1 = bcol1 < N;
        const int  ar  = av  ? arow  : (M - 1);
        const int  bc0 = bv0 ? bcol0 : (N - 1);
        const int  bc1 = bv1 ? bcol1 : (N - 1);
        const int  ka  = K4 + khalf;
        const bool kv0 = ka < K, kv1 = (ka + 1) < K;
        const int  k0  = kv0 ? ka : (K - 1);
        const int  k1  = kv1 ? (ka + 1) : (K - 1);
        v2f a, b0, b1;
        float t0, t1;
        t0 = TRANSA ? A[(long)k0 * lda + ar] : A[(long)ar * lda + k0];
        t1 = TRANSA ? A[(long)k1 * lda + ar] : A[(long)ar * lda + k1];
        a.x = (av && kv0) ? t0 : 0.f;  a.y = (av && kv1) ? t1 : 0.f;
        t0 = TRANSB ? B[(long)bc0 * ldb + k0] : B[(long)k0 * ldb + bc0];
        t1 = TRANSB ? B[(long)bc0 * ldb + k1] : B[(long)k1 * ldb + bc0];
        b0.x = (bv0 && kv0) ? t0 : 0.f;  b0.y = (bv0 && kv1) ? t1 : 0.f;
        t0 = TRANSB ? B[(long)bc1 * ldb + k0] : B[(long)k0 * ldb + bc1];
        t1 = TRANSB ? B[(long)bc1 * ldb + k1] : B[(long)k1 * ldb + bc1];
        b1.x = (bv1 && kv0) ? t0 : 0.f;  b1.y = (bv1 && kv1) ? t1 : 0.f;
        acc0 = wmma_f32(a, b0, acc0);
        acc1 = wmma_f32(a, b1, acc1);
    }

    // ---- epilogue ----
    const int rbase = m0 + ((lane >> 4) << 3);
    #pragma unroll
    for (int t = 0; t < 2; ++t) {
        const int col = (t == 0) ? bcol0 : bcol1;
        v8f&      acc = (t == 0) ? acc0 : acc1;
        if (full || col < N) {
            const float bv = HASBIAS ? bp[col] : 0.f;
            #pragma unroll
            for (int i = 0; i < 8; ++i) {
                const int row = rbase + i;
                if (full || row < M) {
                    float v = acc[i];
                    if (HASBIAS) v += bv;
                    if (ACCUM)   v += Cm[(long)row * ldc + col];
                    if (ACT == 1)      v = tanhf(v);
                    else if (ACT == 2) v = fmaxf(v, 0.f);
                    else if (ACT == 3) v = (row == col) ? -1e9f : fmaxf(v, 0.f);
                    Cm[(long)row * ldc + col] = v;
                }
            }
        }
    }
}

template <bool TA, bool TB, int ACT, bool ACC, bool HB>
static void gemm(hipStream_t s,
                 const float* A, int lda, long long sA,
                 const float* B, int ldb, long long sB,
                 const float* bias, long long sBias,
                 float* C, int ldc, long long sC,
                 int M, int N, int K, int batch)
{
    dim3 grid((N + 31) / 32, (M + 15) / 16, batch);
    gemm_wmma_kernel<TA, TB, ACT, ACC, HB><<<grid, 32, 0, s>>>(
        A, lda, sA, B, ldb, sB, bias, sBias, C, ldc, sC, M, N, K);
}

// ---------------------------------------------------------------------------
// Support kernels
// ---------------------------------------------------------------------------

// DFT tables for rfft(336): dc[t*FF+f]=cos(2pi t f/L), ds=-sin(2pi t f/L)
__global__ void k_dft_tables(float* __restrict__ dc, float* __restrict__ ds)
{
    int i = blockIdx.x * blockDim.x + threadIdx.x;
    if (i >= LL * FF) return;
    int t = i / FF, f = i % FF;
    double ang = -2.0 * 3.14159265358979323846 * (double)(t * f) / (double)LL;
    dc[i] = (float)cos(ang);
    ds[i] = (float)sin(ang);
}

// Instance-norm over L per (b,c); also captures mean/std/last; h -= last.
// x layout (B,L,C); h layout (B,C,L) row index bc=b*CC+c.
__global__ void k_norm(const float* __restrict__ x, const float* __restrict__ gamma,
                       const float* __restrict__ beta, float* __restrict__ h,
                       float* __restrict__ meanb, float* __restrict__ stdb,
                       float* __restrict__ lastb)
{
    const int bc = blockIdx.x;
    const int b = bc / CC, c = bc % CC;
    __shared__ float r1[256], r2[256];
    const int tid = threadIdx.x;
    float s = 0.f, s2 = 0.f;
    for (int l = tid; l < LL; l += 256) {
        float v = x[((long)b * LL + l) * CC + c];
        s += v; s2 += v * v;
    }
    r1[tid] = s; r2[tid] = s2; __syncthreads();
    for (int st = 128; st > 0; st >>= 1) {
        if (tid < st) { r1[tid] += r1[tid + st]; r2[tid] += r2[tid + st]; }
        __syncthreads();
    }
    const float mean = r1[0] / LL;
    float var = r2[0] / LL - mean * mean;
    var = fmaxf(var, 0.f);
    const float sd = sqrtf(var + 1e-5f);
    const float ga = gamma[c], be = beta[c];
    const float last = (x[((long)b * LL + (LL - 1)) * CC + c] - mean) / sd * ga + be;
    if (tid == 0) { meanb[bc] = mean; stdb[bc] = sd; lastb[bc] = last; }
    for (int l = tid; l < LL; l += 256) {
        float v = (x[((long)b * LL + l) * CC + c] - mean) / sd * ga + be;
        h[(long)bc * LL + l] = v - last;
    }
}

// ide[bc,j] = dyn[bc,j] + id_emb[c,j]
__global__ void k_add_rows(float* __restrict__ o, const float* __restrict__ a,
                           const float* __restrict__ rowc, int cols)
{
    long i = (long)blockIdx.x * blockDim.x + threadIdx.x;
    if (i >= (long)BC * cols) return;
    int j = (int)(i % cols);
    int c = (int)((i / cols) % CC);
    o[i] = a[i] + rowc[(long)c * cols + j];
}

__global__ void k_softmax16(float* __restrict__ d)
{
    int r = blockIdx.x * blockDim.x + threadIdx.x;
    if (r >= BC) return;
    float* p = d + (long)r * 16;
    float m = p[0];
    for (int i = 1; i < 16; ++i) m = fmaxf(m, p[i]);
    float s = 0.f;
    for (int i = 0; i < 16; ++i) { float e = expf(p[i] - m); p[i] = e; s += e; }
    float inv = 1.f / s;
    for (int i = 0; i < 16; ++i) p[i] *= inv;
}

// row softmax, one block per row (width = CC)
__global__ void k_softmax_row(float* __restrict__ d, int width)
{
    float* p = d + (long)blockIdx.x * width;
    __shared__ float red[256];
    const int tid = threadIdx.x;
    float m = -3.4e38f;
    for (int i = tid; i < width; i += 256) m = fmaxf(m, p[i]);
    red[tid] = m; __syncthreads();
    for (int st = 128; st > 0; st >>= 1) {
        if (tid < st) red[tid] = fmaxf(red[tid], red[tid + st]);
        __syncthreads();
    }
    m = red[0]; __syncthreads();
    float s = 0.f;
    for (int i = tid; i < width; i += 256) { float e = expf(p[i] - m); p[i] = e; s += e; }
    red[tid] = s; __syncthreads();
    for (int st = 128; st > 0; st >>= 1) {
        if (tid < st) red[tid] += red[tid + st];
        __syncthreads();
    }
    const float inv = 1.f / red[0];
    for (int i = tid; i < width; i += 256) p[i] *= inv;
}

__global__ void k_concat2(float* __restrict__ g, const float* __restrict__ ide,
                          const float* __restrict__ cle)
{
    long i = (long)blockIdx.x * blockDim.x + threadIdx.x;
    if (i >= (long)BC * (IDD + CLD)) return;
    int j = (int)(i % (IDD + CLD));
    long bc = i / (IDD + CLD);
    g[i] = (j < IDD) ? ide[bc * IDD + j] : cle[bc * CLD + (j - IDD)];
}

__global__ void k_concat3(float* __restrict__ gi, const float* __restrict__ t,
                          const float* __restrict__ ide, const float* __restrict__ cle)
{
    long i = (long)blockIdx.x * blockDim.x + threadIdx.x;
    if (i >= (long)BC * GD) return;
    int j = (int)(i % GD);
    long bc = i / GD;
    float v;
    if (j < DD)            v = t[bc * DD + j];
    else if (j < DD + IDD) v = ide[bc * IDD + (j - DD)];
    else                   v = cle[bc * CLD + (j - DD - IDD)];
    gi[i] = v;
}

__global__ void k_copy(float* __restrict__ d, const float* __restrict__ s, long n)
{
    long i = (long)blockIdx.x * blockDim.x + threadIdx.x;
    if (i < n) d[i] = s[i];
}

__global__ void k_addinto(float* __restrict__ d, const float* __restrict__ s, long n)
{
    long i = (long)blockIdx.x * blockDim.x + threadIdx.x;
    if (i < n) d[i] += s[i];
}

// pack (f1im,f2im,f1re,f2re) -> (B,4,C,FF)
__global__ void k_pack_freq(float* __restrict__ A, const float* __restrict__ f1im,
                            const float* __restrict__ f2im, const float* __restrict__ f1re,
                            const float* __restrict__ f2re)
{
    long i = (long)blockIdx.x * blockDim.x + threadIdx.x;
    if (i >= (long)BB * 4 * CC * FF) return;
    int f = (int)(i % FF); long t = i / FF;
    int c = (int)(t % CC); t /= CC;
    int ch = (int)(t % 4); int b = (int)(t / 4);
    long src = ((long)b * CC + c) * FF + f;
    const float* sp = (ch == 0) ? f1im : (ch == 1) ? f2im : (ch == 2) ? f1re : f2re;
    A[i] = sp[src];
}

// fused: pad(1,1) -> conv(1x3) -> +bias -> avgpool(k=2) over W
__global__ void k_conv_pool(const float* __restrict__ in, const float* __restrict__ w,
                            const float* __restrict__ bias, float* __restrict__ out,
                            int cin, int cout, int Win, int Wout)
{
    long idx = (long)blockIdx.x * blockDim.x + threadIdx.x;
    long total = (long)BB * cout * CC * Wout;
    if (idx >= total) return;
    int wo = (int)(idx % Wout); long t = idx / Wout;
    int c  = (int)(t % CC); t /= CC;
    int co = (int)(t % cout); int b = (int)(t / cout);
    const float* wp = w + (long)co * cin * 3;
    float s = 0.f;
    const int x0 = 2 * wo, x1 = 2 * wo + 1;
    for (int ci = 0; ci < cin; ++ci) {
        const float* row = in + (((long)b * cin + ci) * CC + c) * Win;
        const float w0 = wp[ci * 3 + 0], w1 = wp[ci * 3 + 1], w2 = wp[ci * 3 + 2];
        float v0 = ((x0 - 1) >= 0 ? row[x0 - 1] : 0.f) * w0 + row[x0] * w1 +
                   ((x0 + 1) < Win ? row[x0 + 1] : 0.f) * w2;
        float v1 = row[x1 - 1] * w0 + (x1 < Win ? row[x1] : 0.f) * w1 +
                   ((x1 + 1) < Win ? row[x1 + 1] : 0.f) * w2;
        s += v0 + v1;
    }
    out[idx] = 0.5f * s + bias[co];
}

// out[b,p,c] = ((h[b,c,p] + last) - beta)/(gamma+1e-8)*std + mean
__global__ void k_final_out(const float* __restrict__ h, const float* __restrict__ lastb,
                            const float* __restrict__ meanb, const float* __restrict__ stdb,
                            const float* __restrict__ gamma, const float* __restrict__ beta,
                            float* __restrict__ out)
{
    long i = (long)blockIdx.x * blockDim.x + threadIdx.x;
    if (i >= (long)BB * PP * CC) return;
    int c = (int)(i % CC); long t = i / CC;
    int p = (int)(t % PP); int b = (int)(t / PP);
    long bc = (long)b * CC + c;
    float v = h[bc * PP + p] + lastb[bc];
    v = (v - beta[c]) / (gamma[c] + 1e-8f) * stdb[bc] + meanb[bc];
    out[i] = v;
}

// ---------------------------------------------------------------------------
// Parameter bundles + workspace
// ---------------------------------------------------------------------------
struct GmlpP { const float *amp_b, *amp_w, *ph_b, *ph_w, *upd_b, *upd_w; };
struct GcnP  { const float *cluster_emb, *ctx, *gp_b, *gp_w, *i2c_b, *i2c_w, *id_emb,
                           *lin_b, *lin_w, *p1_b, *p1_w, *p2_b, *p2_w; };
struct FconvP { const float* cw[7]; const float* cb[7]; const float *final_b, *final_w; };

struct WS {
    float *dftc, *dfts, *hA, *hB, *meanb, *stdb, *lastb;
    float *t, *xre, *xim, *hmid, *z1, *dyn, *ide, *idc, *logits, *cle,
          *gcat, *gmat, *adj, *gi, *agg, *wctx, *out1, *gx2, *convA, *convB;
};

static const int FC_CIN[7]  = {4, 8, 16, 32, 64, 128, 128};
static const int FC_COUT[7] = {8, 16, 32, 64, 128, 128, 256};
static const int FC_WIN[7]  = {169, 84, 42, 21, 10, 5, 2};
static const int FC_WOUT[7] = {84, 42, 21, 10, 5, 2, 1};

// one model block: gMLP -> midGCN -> freqConv (h updated in place)
static void run_block(hipStream_t s, float* h, const GmlpP& gp, const GcnP& cp,
                      const FconvP& fp, const WS& ws)
{
    // ---------------- gMLP ----------------
    // rfft via DFT matmul: (BC,336) @ (336,169)
    gemm<false, false, 0, false, false>(s, h, LL, 0, ws.dftc, FF, 0, nullptr, 0,
                                        ws.xre, FF, 0, BC, FF, LL, 1);
    gemm<false, false, 0, false, false>(s, h, LL, 0, ws.dfts, FF, 0, nullptr, 0,
                                        ws.xim, FF, 0, BC, FF, LL, 1);
    // per-channel spectral linears (batch=C, M=B): hmid = re@amp_w+amp_b + im@ph_w+ph_b
    gemm<false, false, 0, false, true>(s, ws.xre, CC * FF, FF, gp.amp_w, LL, (long long)FF * LL,
                                       gp.amp_b, LL, ws.hmid, CC * LL, LL, BB, LL, FF, CC);
    gemm<false, false, 0, true, true>(s, ws.xim, CC * FF, FF, gp.ph_w, LL, (long long)FF * LL,
                                      gp.ph_b, LL, ws.hmid, CC * LL, LL, BB, LL, FF, CC);
    // t = tanh(hmid @ upd_w + upd_b)
    gemm<false, false, 1, false, true>(s, ws.hmid, LL, 0, gp.upd_w, DD, 0, gp.upd_b, 0,
                                       ws.t, DD, 0, BC, DD, LL, 1);

    // ---------------- midGCN ----------------
    gemm<false, false, 2, false, true>(s, ws.t, DD, 0, cp.p1_w, DD / 2, 0, cp.p1_b, 0,
                                       ws.z1, DD / 2, 0, BC, DD / 2, DD, 1);
    gemm<false, false, 0, false, true>(s, ws.z1, DD / 2, 0, cp.p2_w, IDD, 0, cp.p2_b, 0,
                                       ws.dyn, IDD, 0, BC, IDD, DD / 2, 1);
    k_add_rows<<<ceil_div((long)BC * IDD, 256), 256, 0, s>>>(ws.ide, ws.dyn, cp.id_emb, IDD);
    gemm<false, false, 0, false, true>(s, ws.ide, IDD, 0, cp.i2c_w, CLD, 0, cp.i2c_b, 0,
                                       ws.idc, CLD, 0, BC, CLD, IDD, 1);
    // logits = idc @ cluster_emb^T ; softmax ; cle = probs @ cluster_emb
    gemm<false, true, 0, false, false>(s, ws.idc, CLD, 0, cp.cluster_emb, CLD, 0, nullptr, 0,
                                       ws.logits, 16, 0, BC, 16, CLD, 1);
    k_softmax16<<<ceil_div(BC, 256), 256, 0, s>>>(ws.logits);
    gemm<false, false, 0, false, false>(s, ws.logits, 16, 0, cp.cluster_emb, CLD, 0, nullptr, 0,
                                        ws.cle, CLD, 0, BC, CLD, 16, 1);
    k_concat2<<<ceil_div((long)BC * (IDD + CLD), 256), 256, 0, s>>>(ws.gcat, ws.ide, ws.cle);
    gemm<false, false, 0, false, true>(s, ws.gcat, IDD + CLD, 0, cp.gp_w, GG, 0, cp.gp_b, 0,
                                       ws.gmat, GG, 0, BC, GG, IDD + CLD, 1);
    // adj scores: off-diag = relu(g.g), diag = -1e9 ; then row softmax
    gemm<false, true, 3, false, false>(s, ws.gmat, GG, (long long)CC * GG,
                                       ws.gmat, GG, (long long)CC * GG, nullptr, 0,
                                       ws.adj, CC, (long long)CC * CC, CC, CC, GG, BB);
    k_softmax_row<<<BC, 256, 0, s>>>(ws.adj, CC);
    k_concat3<<<ceil_div((long)BC * GD, 256), 256, 0, s>>>(ws.gi, ws.t, ws.ide, ws.cle);
    // agg[b] = adj[b] @ gi[b]
    gemm<false, false, 0, false, false>(s, ws.adj, CC, (long long)CC * CC,
                                        ws.gi, GD, (long long)CC * GD, nullptr, 0,
                                        ws.agg, GD, (long long)CC * GD, CC, GD, CC, BB);
    // w[b] = tanh(agg[b]^T @ ctx)
    gemm<true, false, 1, false, false>(s, ws.agg, GD, (long long)CC * GD,
                                       cp.ctx, GD, 0, nullptr, 0,
                                       ws.wctx, GD, (long long)GD * GD, GD, GD, CC, BB);
    // out1[b] = gi[b] + gi[b] @ w[b]
    k_copy<<<ceil_div((long)BC * GD, 256), 256, 0, s>>>(ws.out1, ws.gi, (long)BC * GD);
    gemm<false, false, 0, true, false>(s, ws.gi, GD, (long long)CC * GD,
                                       ws.wctx, GD, (long long)GD * GD, nullptr, 0,
                                       ws.out1, GD, (long long)CC * GD, CC, GD, GD, BB);
    // gx2 = out1 @ lin_w + lin_b
    gemm<false, false, 0, false, true>(s, ws.out1, GD, 0, cp.lin_w, PP, 0, cp.lin_b, 0,
                                       ws.gx2, PP, 0, BC, PP, GD, 1);

    // ---------------- freqConv(h, gx2) ----------------
    float* f1re = ws.xre;  float* f1im = ws.xim;
    float* f2re = ws.t;    float* f2im = ws.t + (size_t)BC * FF;  // t is free now
    gemm<false, false, 0, false, false>(s, h, LL, 0, ws.dftc, FF, 0, nullptr, 0,
                                        f1re, FF, 0, BC, FF, LL, 1);
    gemm<false, false, 0, false, false>(s, h, LL, 0, ws.dfts, FF, 0, nullptr, 0,
                                        f1im, FF, 0, BC, FF, LL, 1);
    gemm<false, false, 0, false, false>(s, ws.gx2, LL, 0, ws.dftc, FF, 0, nullptr, 0,
                                        f2re, FF, 0, BC, FF, LL, 1);
    gemm<false, false, 0, false, false>(s, ws.gx2, LL, 0, ws.dfts, FF, 0, nullptr, 0,
                                        f2im, FF, 0, BC, FF, LL, 1);
    k_pack_freq<<<ceil_div((long)BB * 4 * CC * FF, 256), 256, 0, s>>>(ws.convA, f1im, f2im,
                                                                      f1re, f2re);
    float* ci = ws.convA;
    float* co = ws.convB;
    for (int i = 0; i < 7; ++i) {
        long tot = (long)BB * FC_COUT[i] * CC * FC_WOUT[i];
        k_conv_pool<<<ceil_div(tot, 256), 256, 0, s>>>(ci, fp.cw[i], fp.cb[i], co,
                                                       FC_CIN[i], FC_COUT[i],
                                                       FC_WIN[i], FC_WOUT[i]);
        float* tmp = ci; ci = co; co = tmp;
    }
    // h = x1 + x2 + (final 1x1 conv output, produced directly in (B,C,P) layout)
    k_addinto<<<ceil_div((long)BC * LL, 256), 256, 0, s>>>(h, ws.gx2, (long)BC * LL);
    // out[b,c,p] += sum_s ci[b,s,c] * final_w[p,s] + final_b[p]
    gemm<true, true, 0, true, true>(s, ci, CC, (long long)256 * CC,
                                    fp.final_w, 256, 0, fp.final_b, 0,
                                    h, LL, (long long)CC * LL, CC, PP, 256, BB);
}

// ---------------------------------------------------------------------------
// Entry point.  Leaf order assumes JAX pytree flattening of setup_inputs():
// dicts sorted by key, tuples in order -> params leaves first, then x.
// ---------------------------------------------------------------------------
extern "C" void kernel_launch(void* const* d_in, const int* in_sizes, int n_in,
                              void* d_out, int out_size, void* d_ws, size_t ws_size,
                              hipStream_t stream)
{
    (void)in_sizes; (void)n_in; (void)out_size; (void)ws_size;

    int idx = 0;
    auto nx = [&]() { return (const float*)d_in[idx++]; };

    const float* beta_p   = nx();   // params.beta       (1,C,1,1)
    const float* fcidp_b  = nx();   // params.fc_idp_b   (C,1,P)
    const float* fcidp_w  = nx();   // params.fc_idp_w   (C,1,L,P)
    FconvP fin, fout;
    for (int i = 0; i < 7; ++i) { fin.cw[i] = nx(); fin.cb[i] = nx(); }
    fin.final_b = nx(); fin.final_w = nx();
    for (int i = 0; i < 7; ++i) { fout.cw[i] = nx(); fout.cb[i] = nx(); }
    fout.final_b = nx(); fout.final_w = nx();
    const float* gamma_p = nx();    // params.gamma
    GcnP gin[2], gcn_out;
    auto read_gcn = [&](GcnP& g) {
        g.cluster_emb = nx(); g.ctx = nx(); g.gp_b = nx(); g.gp_w = nx();
        g.i2c_b = nx(); g.i2c_w = nx(); g.id_emb = nx(); g.lin_b = nx();
        g.lin_w = nx(); g.p1_b = nx(); g.p1_w = nx(); g.p2_b = nx(); g.p2_w = nx();
    };
    read_gcn(gin[0]); read_gcn(gin[1]); read_gcn(gcn_out);
    GmlpP tin[2], tmlp_out;
    auto read_gmlp = [&](GmlpP& g) {
        g.amp_b = nx(); g.amp_w = nx(); g.ph_b = nx(); g.ph_w = nx();
        g.upd_b = nx(); g.upd_w = nx();
    };
    read_gmlp(tin[0]); read_gmlp(tin[1]); read_gmlp(tmlp_out);
    const float* x = nx();          // x (B,L,C)

    // ---- workspace (bump allocator, fp32) ----
    float* base = (float*)d_ws;
    size_t off = 0;
    auto al = [&](size_t n) { float* p = base + off; off += n; return p; };
    WS ws;
    ws.dftc  = al((size_t)LL * FF);
    ws.dfts  = al((size_t)LL * FF);
    ws.hA    = al((size_t)BC * LL);
    ws.hB    = al((size_t)BC * LL);
    ws.meanb = al(BC); ws.stdb = al(BC); ws.lastb = al(BC);
    ws.t     = al((size_t)BC * DD);          // also reused for f2re/f2im
    ws.xre   = al((size_t)BC * FF);
    ws.xim   = al((size_t)BC * FF);
    ws.hmid  = al((size_t)BC * LL);
    ws.z1    = al((size_t)BC * (DD / 2));
    ws.dyn   = al((size_t)BC * IDD);
    ws.ide   = al((size_t)BC * IDD);
    ws.idc   = al((size_t)BC * CLD);
    ws.logits= al((size_t)BC * 16);
    ws.cle   = al((size_t)BC * CLD);
    ws.gcat  = al((size_t)BC * (IDD + CLD));
    ws.gmat  = al((size_t)BC * GG);
    ws.adj   = al((size_t)BB * CC * CC);
    ws.gi    = al((size_t)BC * GD);
    ws.agg   = al((size_t)BC * GD);
    ws.wctx  = al((size_t)BB * GD * GD);
    ws.out1  = al((size_t)BC * GD);
    ws.gx2   = al((size_t)BC * PP);
    ws.convA = al((size_t)1736064);          // max stage tensor: B*4*C*169
    ws.convB = al((size_t)1736064);

    // DFT tables + instance norm
    k_dft_tables<<<ceil_div((long)LL * FF, 256), 256, 0, stream>>>(ws.dftc, ws.dfts);
    k_norm<<<BC, 256, 0, stream>>>(x, gamma_p, beta_p, ws.hA, ws.meanb, ws.stdb, ws.lastb);

    float* h = ws.hA;
    float* hspare = ws.hB;

    // two encoder blocks (shared fconv_in params)
    for (int l = 0; l < 2; ++l)
        run_block(stream, h, tin[l], gin[l], fin, ws);

    // fc_idp: per-channel (B,L)@(L,P)+b  -> hspare, then swap
    gemm<false, false, 0, false, true>(stream, h, CC * LL, LL,
                                       fcidp_w, PP, (long long)LL * PP,
                                       fcidp_b, PP,
                                       hspare, CC * PP, PP, BB, PP, LL, CC);
    { float* tmp = h; h = hspare; hspare = tmp; }

    // output block
    run_block(stream, h, tmlp_out, gcn_out, fout, ws);

    // add last back, de-normalize, transpose to (B,P,C)
    k_final_out<<<ceil_div((long)BB * PP * CC, 256), 256, 0, stream>>>(
        h, ws.lastb, ws.meanb, ws.stdb, gamma_p, beta_p, (float*)d_out);
}